// FWIForward_6150393168425
// MI455X (gfx1250) — compile-verified
//
#include <hip/hip_runtime.h>
#include <cmath>

#ifndef M_PI
#define M_PI 3.14159265358979323846
#endif

#define NXg    360
#define N2g    (NXg * NXg)
#define NTRg   32
#define NMEASg 256
#define NSTEPSg 640
#define NBCg   15
#define PADg   73
#define NVIN   214

#define TILE   16
#define HALO   2
#define HT     (TILE + 2 * HALO)          // 20
#define NCHUNK (HT / 4)                   // 5 four-cell chunks per halo row
#define NTILES ((NXg + TILE - 1) / TILE)  // 23

#if defined(__AMDGCN__)
typedef int v4i_t __attribute__((ext_vector_type(4)));
typedef __attribute__((address_space(1))) int   g_int;
typedef __attribute__((address_space(3))) int   l_int;
typedef __attribute__((address_space(1))) v4i_t g_v4i;
typedef __attribute__((address_space(3))) v4i_t l_v4i;
#endif

__device__ __forceinline__ void async_copy_b32(const float* gp, float* lp) {
#if defined(__gfx1250__) && __has_builtin(__builtin_amdgcn_global_load_async_to_lds_b32)
    __builtin_amdgcn_global_load_async_to_lds_b32((g_int*)gp, (l_int*)lp, 0, 0);
#else
    unsigned loff = (unsigned)(unsigned long long)lp;
    asm volatile("global_load_async_to_lds_b32 %0, %1, off"
                 :: "v"(loff), "v"(gp) : "memory");
#endif
}

#if defined(__gfx1250__) && __has_builtin(__builtin_amdgcn_global_load_async_to_lds_b128)
#define HAVE_ASYNC_B128 1
__device__ __forceinline__ void async_copy_b128(const float* gp, float* lp) {
    __builtin_amdgcn_global_load_async_to_lds_b128((g_v4i*)gp, (l_v4i*)lp, 0, 0);
}
#endif

__device__ __forceinline__ void async_wait_all() {
#if defined(__gfx1250__) && __has_builtin(__builtin_amdgcn_s_wait_asynccnt)
    __builtin_amdgcn_s_wait_asynccnt(0);
#else
    asm volatile("s_wait_asynccnt 0" ::: "memory");
#endif
}

// ---------------- main stencil step (placed first so the disasm snippet shows it) ----
// Internal field layout: p[cell][trace], cell = x*NX + y, trace = lane (wave32 == NTR).

__global__ __launch_bounds__(256) void step_kernel(
    const float* __restrict__ p1,    // current field (staged to LDS; also gathered for meas)
    float* p0,                       // previous field (read once per cell) and write target
    const float4* __restrict__ coef, // {alpha, temp1, temp0, -} per cell
    const int* __restrict__ in_inds, const float* __restrict__ beta,
    const int* __restrict__ meas_inds, float* __restrict__ out,
    int out_step, float srcval)
{
    const float C2 = 4.0f / 3.0f, C3 = -1.0f / 12.0f;
    __shared__ float sm[HT * HT * NTRg];   // 20*20*32*4B = 51200 B
    const int tid = threadIdx.x;
    const int wave = tid >> 5, lane = tid & 31;
    const int x0 = blockIdx.x * TILE, y0 = blockIdx.y * TILE;

    // Stage (TILE+4)^2 halo tile of p1 into LDS.
    // Fast path: one async b128 per 4-cell chunk (512B/instruction). Rows never
    // straddle a chunk (HT % 4 == 0); only y-wrapping chunks fall back to b32.
    for (int k = wave; k < HT * NCHUNK; k += 8) {
        int r = k / NCHUNK, kc = k - r * NCHUNK;
        int gx = x0 + r - HALO; if (gx < 0) gx += NXg; else if (gx >= NXg) gx -= NXg;
        int gy0 = y0 + kc * 4 - HALO;
        float* lp = &sm[(r * HT + kc * 4) * NTRg];
#if defined(HAVE_ASYNC_B128)
        if (gy0 >= 0 && gy0 + 3 < NXg) {
            const float* gp = p1 + (size_t)(gx * NXg + gy0) * NTRg + lane * 4;
            async_copy_b128(gp, lp + lane * 4);
            continue;
        }
#endif
        for (int t = 0; t < 4; ++t) {
            int gy = gy0 + t; if (gy < 0) gy += NXg; else if (gy >= NXg) gy -= NXg;
            const float* gp = p1 + (size_t)(gx * NXg + gy) * NTRg + lane;
            async_copy_b32(gp, lp + t * NTRg + lane);
        }
    }
    async_wait_all();
    __syncthreads();

    const int   my_in  = in_inds[lane];
    const float my_inj = beta[lane] * srcval;

    for (int c = wave; c < TILE * TILE; c += 8) {
        int i = c >> 4, j = c & (TILE - 1);
        int gx = x0 + i, gy = y0 + j;
        if (gx >= NXg || gy >= NXg) continue;
        int b = ((i + HALO) * HT + (j + HALO)) * NTRg + lane;
        float cc = sm[b];
        float l1 = sm[b - HT * NTRg] + sm[b + HT * NTRg] + sm[b - NTRg] + sm[b + NTRg];
        float l2 = sm[b - 2 * HT * NTRg] + sm[b + 2 * HT * NTRg]
                 + sm[b - 2 * NTRg] + sm[b + 2 * NTRg];
        float lap = C2 * l1 + C3 * l2;
        int g = gx * NXg + gy;
        size_t gi = (size_t)g * NTRg + lane;
        float4 cf = coef[g];                         // wave-uniform b128 load
        float pn = cf.x * lap + cf.y * cc - cf.z * p0[gi];
        if (g == my_in) pn += my_inj;                // per-trace source injection
        p0[gi] = pn;                                 // in-place: p0[cell] already consumed
    }

    // Fused receiver gather: p1 is the (post-injection) field of step out_step and
    // is read-only during this kernel; one block handles all 256x32 samples.
    if (out_step >= 0 && blockIdx.x == 0 && blockIdx.y == 0) {
        for (int k = tid; k < NTRg * NMEASg; k += 256) {
            int tr = k >> 8, j = k & (NMEASg - 1);
            out[((size_t)tr * NSTEPSg + out_step) * NMEASg + j] =
                p1[(size_t)meas_inds[j] * NTRg + tr];
        }
    }
}

// Standalone gather for the final step.
__global__ void measure_kernel(const float* __restrict__ p, const int* __restrict__ meas_inds,
                               float* __restrict__ out, int step) {
    int tr = blockIdx.x, j = threadIdx.x;
    out[((size_t)tr * NSTEPSg + step) * NMEASg + j] = p[(size_t)meas_inds[j] * NTRg + tr];
}

// ---------------- init kernels ----------------

__global__ void init_inds_kernel(int* __restrict__ meas_inds, int* __restrict__ in_inds,
                                 int* __restrict__ velmin_bits) {
    int m = threadIdx.x;
    if (m == 0) *velmin_bits = __float_as_int(3.0e38f);
    double theta = 2.0 * M_PI * (double)m / (double)NMEASg;
    int t0 = (int)floor(160.0 * cos(theta) + (double)(NXg - 1) * 0.5);
    int t1 = (int)floor(160.0 * sin(theta) + (double)(NXg - 1) * 0.5);
    int cell = NXg * t0 + t1;
    meas_inds[m] = cell;
    if ((m & 7) == 0) in_inds[m >> 3] = cell;   // IN_INDS = MEAS_INDS[::8]
}

__global__ void pad_kernel(const float* __restrict__ vin, float* __restrict__ vpad,
                           int* __restrict__ velmin_bits) {
    __shared__ float smin[256];
    int idx = blockIdx.x * 256 + threadIdx.x;
    float val = 3.0e38f;
    if (idx < N2g) {
        int x = idx / NXg, y = idx - x * NXg;
        val = 1.5f;
        if (x >= PADg && x < PADg + NVIN && y >= PADg && y < PADg + NVIN)
            val = vin[(x - PADg) * NVIN + (y - PADg)];
        vpad[idx] = val;
    }
    smin[threadIdx.x] = val;
    __syncthreads();
    for (int s = 128; s > 0; s >>= 1) {
        if (threadIdx.x < s) smin[threadIdx.x] = fminf(smin[threadIdx.x], smin[threadIdx.x + s]);
        __syncthreads();
    }
    if (threadIdx.x == 0) atomicMin(velmin_bits, __float_as_int(smin[0]));  // all values > 0
}

__global__ void coeff_kernel(const float* __restrict__ vpad, const int* __restrict__ velmin_bits,
                             float4* __restrict__ coef) {
    int idx = blockIdx.x * 256 + threadIdx.x;
    if (idx >= N2g) return;
    const float DT = 0.2f, DX = 0.6f, C1 = -2.5f;
    int x = idx / NXg, y = idx - x * NXg;
    float velmin = __int_as_float(*velmin_bits);
    const float a = (NBCg - 1) * DX;                        // 8.4
    float kap = 3.0f * velmin * logf(10000.0f) / (2.0f * a);
    // overwrite priority from reference: right cols > left cols > bottom rows > top rows
    float i, d;
    if (y >= NXg - NBCg)      { i = (float)(y - (NXg - NBCg)); d = kap * (i * DX / a) * (i * DX / a); }
    else if (y < NBCg)        { i = (float)(NBCg - 1 - y);     d = kap * (i * DX / a) * (i * DX / a); }
    else if (x >= NXg - NBCg) { i = (float)(x - (NXg - NBCg)); d = kap * (i * DX / a) * (i * DX / a); }
    else if (x < NBCg)        { i = (float)(NBCg - 1 - x);     d = kap * (i * DX / a) * (i * DX / a); }
    else d = 0.0f;
    float kappa = d * DT;
    float v = vpad[idx], v2 = v * v;
    float al = v2 * (DT / DX) * (DT / DX);
    float4 cf;
    cf.x = al;                             // alpha
    cf.y = 2.0f + 2.0f * C1 * al - kappa;  // temp1
    cf.z = 1.0f - kappa;                   // temp0
    cf.w = 0.0f;
    coef[idx] = cf;
}

__global__ void beta_kernel(const float* __restrict__ vpad, const int* __restrict__ in_inds,
                            float* __restrict__ beta) {
    int tr = threadIdx.x;
    if (tr >= NTRg) return;
    float v = vpad[in_inds[tr]];
    beta[tr] = 0.2f * 0.2f * v * v;
}

// ---------------- host ----------------

extern "C" void kernel_launch(void* const* d_in, const int* in_sizes, int n_in,
                              void* d_out, int out_size, void* d_ws, size_t ws_size,
                              hipStream_t stream) {
    const float* vin = (const float*)d_in[0];
    float* out = (float*)d_out;

    size_t fieldN = (size_t)N2g * NTRg;
    float*  bufA      = (float*)d_ws;             // field buffer 0
    float*  bufB      = bufA + fieldN;            // field buffer 1
    float4* coef      = (float4*)(bufB + fieldN); // 16B-aligned (prefix is multiple of 16B)
    float*  vpad      = (float*)(coef + N2g);
    int*    meas_inds = (int*)(vpad + N2g);
    int*    in_inds   = meas_inds + NMEASg;
    float*  beta      = (float*)(in_inds + NTRg);
    int*    velmin_b  = (int*)(beta + NTRg);

    (void)hipMemsetAsync(bufA, 0, 2 * fieldN * sizeof(float), stream);
    init_inds_kernel<<<1, NMEASg, 0, stream>>>(meas_inds, in_inds, velmin_b);
    pad_kernel<<<(N2g + 255) / 256, 256, 0, stream>>>(vin, vpad, velmin_b);
    coeff_kernel<<<(N2g + 255) / 256, 256, 0, stream>>>(vpad, velmin_b, coef);
    beta_kernel<<<1, NTRg, 0, stream>>>(vpad, in_inds, beta);

    dim3 grid(NTILES, NTILES);
    for (int s = 0; s < NSTEPSg; ++s) {
        double t = (double)s * 0.2;
        float src = (float)(exp(-(t - 3.2) * (t - 3.2) / (2.0 * 12.0 * 12.0))
                            * sin(2.0 * M_PI * t));
        float* pw = (s & 1) ? bufB : bufA;        // p0 (read-once) and write target
        const float* pc = (s & 1) ? bufA : bufB;  // p1 (stencil source, field of step s-1)
        step_kernel<<<grid, 256, 0, stream>>>(pc, pw, coef, in_inds, beta,
                                              meas_inds, out, s - 1, src);
    }
    // field of the final step lives in bufB (NSTEPS-1 is odd)
    measure_kernel<<<NTRg, NMEASg, 0, stream>>>(bufB, meas_inds, out, NSTEPSg - 1);
}